// ConvertParamMomentsTEtoParamsTE_76596446757545
// MI455X (gfx1250) — compile-verified
//
#include <hip/hip_runtime.h>

// ---------------------------------------------------------------------------
// ConvertParamMomentsTEtoParamsTE  (MI455X / gfx1250)
//
// Pure streaming problem: 2 x 128MB f32 matrices read once, 1 x 128MB written
// once, tiny vector math. HBM-bound (~384MB @ 23.3TB/s ~= 16.5us floor).
//  - one fused pass over varvhTE/varvh, b128 non-temporal loads/stores
//  - CDNA5 async global->LDS staging of per-row coefficients (+s_wait_asynccnt)
//  - deterministic two-stage reductions through workspace (no float atomics)
//  - bTE chunk-reduction done on the matrix pipe: ones[1x64] @ bPart[64xNV]
//    via V_WMMA_F32_16X16X4_F32 (A = all-ones, accumulate over K)
// ---------------------------------------------------------------------------

#define NV 8192
#define NH 4096
#define TPB 256
#define COLS_PER_THREAD 4
#define COLS_PER_BLOCK (TPB * COLS_PER_THREAD)   // 1024
#define GRIDX (NV / COLS_PER_BLOCK)              // 8 column tiles

typedef float v2f __attribute__((ext_vector_type(2)));
typedef float v4f __attribute__((ext_vector_type(4)));
typedef float v8f __attribute__((ext_vector_type(8)));

#define AS1 __attribute__((address_space(1)))
#define AS3 __attribute__((address_space(3)))

#if defined(__has_builtin)
#if __has_builtin(__builtin_amdgcn_global_load_async_to_lds_b32)
#define USE_ASYNC_LDS 1
#endif
#if __has_builtin(__builtin_amdgcn_wmma_f32_16x16x4_f32)
#define USE_WMMA_REDUCE 1
#endif
#endif
#ifndef USE_ASYNC_LDS
#define USE_ASYNC_LDS 0
#endif
#ifndef USE_WMMA_REDUCE
#define USE_WMMA_REDUCE 0
#endif

// ---------------------------------------------------------------------------
// Kernel 0: per-row coefficient vectors (all O(NH) work) + pass-through copies
//   coeff[0*NH+h] = c1  = inv*muh            (weight on varvhTE in bTE, negated)
//   coeff[1*NH+h] = c2  = d*muh - inv*muhTE  (weight on varvh   in bTE)
//   coeff[2*NH+h] = d   = inv*varh_diagTE*inv
//   coeff[3*NH+h] = inv = 1/varh_diag
// ---------------------------------------------------------------------------
__global__ void prep_kernel(const float* __restrict__ varh_diagTE,
                            const float* __restrict__ varh_diag,
                            const float* __restrict__ muh,
                            const float* __restrict__ muhTE,
                            float* __restrict__ coeff,
                            float* __restrict__ out_muhTE,
                            float* __restrict__ out_vdTE) {
  int h = blockIdx.x * blockDim.x + threadIdx.x;
  if (h < NH) {
    float iv  = 1.0f / varh_diag[h];
    float vte = varh_diagTE[h];
    float mte = muhTE[h];
    float d   = iv * vte * iv;
    coeff[0 * NH + h] = iv * muh[h];
    coeff[1 * NH + h] = d * muh[h] - iv * mte;
    coeff[2 * NH + h] = d;
    coeff[3 * NH + h] = iv;
    out_muhTE[h] = mte;   // pass-through output 4
    out_vdTE[h]  = vte;   // pass-through output 5
  }
}

// ---------------------------------------------------------------------------
// Kernel 1: fused streaming pass (the 384MB of traffic lives here).
// grid = (GRIDX column tiles, nch row chunks). Each thread owns 4 consecutive
// columns and walks rowsPerChunk rows:
//   wtTE[h,v]   = inv[h]*varvhTE[h,v] - d[h]*varvh[h,v]      (NT store)
//   bPart[c,v] += c2[h]*varvh[h,v]    - c1[h]*varvhTE[h,v]   (registers)
//   trPart     += 2*inv[h]*varvhTE[h,v]*varvh[h,v] - d[h]*varvh[h,v]^2
// ---------------------------------------------------------------------------
__global__ void fused_main_kernel(const float* __restrict__ varvhTE,
                                  const float* __restrict__ varvh,
                                  float* __restrict__ coeff,     // [4*NH]
                                  float* __restrict__ wt_out,    // [NH*NV]
                                  float* __restrict__ bPart,     // [nch*NV]
                                  float* __restrict__ trPart,    // [nch*GRIDX]
                                  int rowsPerChunk) {
  extern __shared__ float sc[];                 // 4 * rowsPerChunk floats
  __shared__ float red[TPB];

  const int tid   = threadIdx.x;
  const int chunk = blockIdx.y;
  const int h0    = chunk * rowsPerChunk;
  const int col0  = blockIdx.x * COLS_PER_BLOCK + tid * COLS_PER_THREAD;
  const int total = 4 * rowsPerChunk;

#if USE_ASYNC_LDS
  for (int t = tid; t < total; t += TPB) {
    int a = t / rowsPerChunk;
    int i = t - a * rowsPerChunk;
    __builtin_amdgcn_global_load_async_to_lds_b32(
        (AS1 int*)(coeff + (size_t)a * NH + h0 + i),
        (AS3 int*)(sc + t), 0, 0);
  }
  asm volatile("s_wait_asynccnt 0" ::: "memory");
  __syncthreads();
#else
  for (int t = tid; t < total; t += TPB) {
    int a = t / rowsPerChunk;
    int i = t - a * rowsPerChunk;
    sc[t] = coeff[(size_t)a * NH + h0 + i];
  }
  __syncthreads();
#endif

  const float* __restrict__ sc1  = sc;
  const float* __restrict__ sc2  = sc + rowsPerChunk;
  const float* __restrict__ sd   = sc + 2 * rowsPerChunk;
  const float* __restrict__ sinv = sc + 3 * rowsPerChunk;

  v4f   accb  = {0.f, 0.f, 0.f, 0.f};
  float acctr = 0.f;

  size_t base = (size_t)h0 * NV + col0;
#pragma unroll 4
  for (int r = 0; r < rowsPerChunk; ++r) {
    const v4f te = __builtin_nontemporal_load((const v4f*)(varvhTE + base));
    const v4f vv = __builtin_nontemporal_load((const v4f*)(varvh + base));
    const float c1 = sc1[r];
    const float c2 = sc2[r];
    const float d  = sd[r];
    const float iv = sinv[r];

    v4f wt = iv * te - d * vv;
    __builtin_nontemporal_store(wt, (v4f*)(wt_out + base));

    accb += c2 * vv - c1 * te;

    v4f tt = (2.0f * iv) * (te * vv) - d * (vv * vv);
    acctr += (tt.x + tt.y) + (tt.z + tt.w);

    base += NV;
  }

  // bTE partial for this row chunk (one owner per column -> plain store)
  *(v4f*)(bPart + (size_t)chunk * NV + col0) = accb;

  // deterministic in-block trace reduction
  red[tid] = acctr;
  __syncthreads();
  for (int s = TPB / 2; s > 0; s >>= 1) {
    if (tid < s) red[tid] += red[tid + s];
    __syncthreads();
  }
  if (tid == 0) trPart[chunk * GRIDX + blockIdx.x] = red[0];
}

// ---------------------------------------------------------------------------
// Kernel 2a: bTE = muvTE + ones[1 x nch] @ bPart[nch x NV], on the WMMA pipe.
// One wave per 16-column group (NV/16 = 512 groups; 64 blocks x 8 waves).
// V_WMMA_F32_16X16X4_F32: A = 16x4 all-ones (both A VGPRs = 1.0 in every
// lane), B = 4-chunk x 16-column tile:
//   B VGPR0: lanes 0-15 -> K=0, lanes 16-31 -> K=2
//   B VGPR1: lanes 0-15 -> K=1, lanes 16-31 -> K=3
// => lane loads bPart[(c + 2*(lane>>4) + {0,1})*NV + n0 + (lane&15)].
// Every row of D holds the column sums; D VGPR0 lanes 0-15 (M=0,N=0-15) is
// read out. EXEC is all-ones throughout (uniform control flow; writes are
// predicated only after the WMMA chain).
// ---------------------------------------------------------------------------
__global__ void bte_reduce_kernel(const float* __restrict__ muvTE,
                                  const float* __restrict__ bPart,
                                  float* __restrict__ bTE_out,
                                  int nch) {
  const int lane = threadIdx.x & 31;
  const int wid  = threadIdx.x >> 5;
  const int n0   = (blockIdx.x * (TPB / 32) + wid) * 16;  // 16-column group
  const int half = lane >> 4;        // 0: K = 0,1; 1: K = 2,3
  const int nidx = lane & 15;        // column within group

  float colsum;
#if USE_WMMA_REDUCE
  const int nch4 = nch & ~3;
  v8f acc = {0.f, 0.f, 0.f, 0.f, 0.f, 0.f, 0.f, 0.f};
  v2f ones = {1.f, 1.f};
  for (int c = 0; c < nch4; c += 4) {
    v2f b;
    b.x = bPart[(size_t)(c + 2 * half + 0) * NV + n0 + nidx];  // K = 0 / 2
    b.y = bPart[(size_t)(c + 2 * half + 1) * NV + n0 + nidx];  // K = 1 / 3
    acc = __builtin_amdgcn_wmma_f32_16x16x4_f32(
        /*neg_a=*/false, ones, /*neg_b=*/false, b,
        /*c_mod=*/(short)0, acc, /*reuse_a=*/false, /*reuse_b=*/false);
  }
  colsum = acc[0];                       // M=0 row: lanes 0-15 = N=0..15
  for (int c = nch4; c < nch; ++c)       // remainder chunks (nch<4 case only)
    colsum += bPart[(size_t)c * NV + n0 + nidx];
#else
  colsum = 0.f;
  for (int c = 0; c < nch; ++c)
    colsum += bPart[(size_t)c * NV + n0 + nidx];
#endif

  if (lane < 16) bTE_out[n0 + nidx] = muvTE[n0 + nidx] + colsum;
}

// ---------------------------------------------------------------------------
// Kernel 2b: trace reduction -> sig2TE (single block, deterministic order).
// ---------------------------------------------------------------------------
__global__ void sig2_kernel(const float* __restrict__ trPart,
                            const float* __restrict__ varvbarTE,
                            float* __restrict__ sig2_out, int ntr) {
  __shared__ float red[TPB];
  float t = 0.f;
  for (int i = threadIdx.x; i < ntr; i += TPB) t += trPart[i];
  red[threadIdx.x] = t;
  __syncthreads();
  for (int s = TPB / 2; s > 0; s >>= 1) {
    if (threadIdx.x < s) red[threadIdx.x] += red[threadIdx.x + s];
    __syncthreads();
  }
  if (threadIdx.x == 0)
    sig2_out[0] = (varvbarTE[0] - red[0]) / (float)NV;
}

// ---------------------------------------------------------------------------
extern "C" void kernel_launch(void* const* d_in, const int* in_sizes, int n_in,
                              void* d_out, int out_size, void* d_ws, size_t ws_size,
                              hipStream_t stream) {
  (void)in_sizes; (void)n_in; (void)out_size;

  const float* muvTE       = (const float*)d_in[0];
  const float* varvhTE     = (const float*)d_in[1];
  const float* varh_diagTE = (const float*)d_in[2];
  const float* varh_diag   = (const float*)d_in[3];
  const float* muh         = (const float*)d_in[4];
  const float* varvh       = (const float*)d_in[5];
  const float* muhTE       = (const float*)d_in[6];
  const float* varvbarTE   = (const float*)d_in[7];

  // output layout (flat, return order): bTE | wtTE | sig2TE | muhTE | varh_diagTE
  float* out       = (float*)d_out;
  float* bTE_out   = out;
  float* wt_out    = out + NV;
  float* sig2_out  = wt_out + (size_t)NH * NV;
  float* muhTE_out = sig2_out + 1;
  float* vdTE_out  = muhTE_out + NH;

  // workspace layout: coeff[4*NH] | bPart[nch*NV] | trPart[nch*GRIDX]
  float* coeff = (float*)d_ws;
  size_t used  = 4 * (size_t)NH * sizeof(float);
  int nch = 64;  // 64 row chunks x 8 column tiles = 512 blocks
  while (nch > 1 &&
         used + ((size_t)nch * NV + (size_t)nch * GRIDX) * sizeof(float) > ws_size)
    nch >>= 1;
  float* bPart  = (float*)((char*)d_ws + used);
  float* trPart = bPart + (size_t)nch * NV;
  const int rows = NH / nch;

  prep_kernel<<<(NH + TPB - 1) / TPB, TPB, 0, stream>>>(
      varh_diagTE, varh_diag, muh, muhTE, coeff, muhTE_out, vdTE_out);

  dim3 grid1(GRIDX, nch);
  size_t shmem = 4 * (size_t)rows * sizeof(float);
  fused_main_kernel<<<grid1, TPB, shmem, stream>>>(
      varvhTE, varvh, coeff, wt_out, bPart, trPart, rows);

  // NV/16 column groups, one wave each: 512 / (256/32) = 64 blocks
  bte_reduce_kernel<<<(NV / 16) / (TPB / 32), TPB, 0, stream>>>(
      muvTE, bPart, bTE_out, nch);

  sig2_kernel<<<1, TPB, 0, stream>>>(trPart, varvbarTE, sig2_out, nch * GRIDX);
}